// GCN_two_layers_19602230739361
// MI455X (gfx1250) — compile-verified
//
#include <hip/hip_runtime.h>

typedef __attribute__((ext_vector_type(2))) float v2f;
typedef __attribute__((ext_vector_type(8))) float v8f;

#define HAS_WMMA_F32X4 __has_builtin(__builtin_amdgcn_wmma_f32_16x16x4_f32)

// ---------------------------------------------------------------------------
// init: deg = 1.0 (self loop), agg1 = 0 (vectorized), agg2 = 0
// grid covers N*4 float4 slots of agg1
// ---------------------------------------------------------------------------
__global__ __launch_bounds__(256) void init_ws_kernel(float* __restrict__ deg,
                                                      float4* __restrict__ agg1,
                                                      float* __restrict__ agg2,
                                                      int N) {
  int i = blockIdx.x * blockDim.x + threadIdx.x;
  int n4 = N * 4;
  if (i < n4) agg1[i] = make_float4(0.f, 0.f, 0.f, 0.f);
  if (i < N) { deg[i] = 1.0f; agg2[i] = 0.0f; }
}

// ---------------------------------------------------------------------------
// degree accumulation over dst (self-loops already counted via init=1)
// ---------------------------------------------------------------------------
__global__ __launch_bounds__(256) void deg_accum_kernel(const long long* __restrict__ dst,
                                                        float* __restrict__ deg, int E) {
  int e = blockIdx.x * blockDim.x + threadIdx.x;
  if (e >= E) return;
  atomicAdd(&deg[(int)dst[e]], 1.0f);
}

// deg -> deg^{-1/2} in place (deg >= 1 always, self-loops guarantee it)
__global__ __launch_bounds__(256) void dinv_kernel(float* __restrict__ deg, int N) {
  int n = blockIdx.x * blockDim.x + threadIdx.x;
  if (n >= N) return;
  deg[n] = __frsqrt_rn(deg[n]);
}

// ---------------------------------------------------------------------------
// h1 = x @ W1   (N x 3) @ (3 x 16), K padded to 4 with zeros.
// One wave per 16-row tile; V_WMMA_F32_16X16X4_F32.
//   A 16x4 layout : lanes 0-15 -> {K0,K1}, lanes 16-31 -> {K2,K3}, M = lane&15
//   B 4x16 layout : mirrored (N = lane&15)
//   D 16x16 layout: VGPR r -> row r + 8*(lane>>4), col = lane&15
// Full-tile fast path: wave-uniform branch, unconditional stores, clamped
// loads + v_cndmask for the K=3 zero padding (no per-lane exec branches).
// ---------------------------------------------------------------------------
__global__ __launch_bounds__(256) void h1_wmma_kernel(const float* __restrict__ x,
                                                      const float* __restrict__ W1,
                                                      float* __restrict__ h1, int N) {
  int wave = (blockIdx.x * blockDim.x + threadIdx.x) >> 5;
  int lane = threadIdx.x & 31;
  int row0 = wave * 16;
  if (row0 >= N) return;            // uniform per wave -> EXEC stays all ones

  int m  = lane & 15;
  int kh = lane >> 4;               // 0 or 1
  int k0 = 2 * kh;                  // K of first element: 0 or 2 (always < 3)
  int row = row0 + m;
  bool full = (row0 + 16 <= N);     // wave-uniform

  if (full) {
#if HAS_WMMA_F32X4
    v2f a, b;
    a.x = x[row * 3 + k0];                         // K=0 or K=2, valid
    // K=1 (kh==0) or zero-pad K=3 (kh==1): load from safe clamped address,
    // select with cndmask instead of exec branch.
    {
      int ai = row * 3 + ((kh == 0) ? 1 : 2);      // always in bounds
      float av = x[ai];
      a.y = (kh == 0) ? av : 0.0f;
    }
    b.x = W1[k0 * 16 + m];                         // row 0 or row 2 of W1
    {
      int bi = ((kh == 0) ? 16 : 0) + m;           // row 1 of W1 (or safe row 0)
      float bv = W1[bi];
      b.y = (kh == 0) ? bv : 0.0f;                 // zero-pad K=3
    }
    v8f c = {};
    c = __builtin_amdgcn_wmma_f32_16x16x4_f32(false, a, false, b, (short)0, c,
                                              false, false);
#pragma unroll
    for (int r = 0; r < 8; ++r)
      h1[(row0 + r + 8 * kh) * 16 + m] = c[r];
#else
#pragma unroll
    for (int r = 0; r < 8; ++r) {
      int orow = row0 + r + 8 * kh;
      float acc = 0.0f;
#pragma unroll
      for (int k = 0; k < 3; ++k) acc += x[orow * 3 + k] * W1[k * 16 + m];
      h1[orow * 16 + m] = acc;
    }
#endif
  } else {
    // boundary tile: guarded scalar path (runs at most for one wave)
#pragma unroll
    for (int r = 0; r < 8; ++r) {
      int orow = row0 + r + 8 * kh;
      if (orow < N) {
        float acc = 0.0f;
#pragma unroll
        for (int k = 0; k < 3; ++k) acc += x[orow * 3 + k] * W1[k * 16 + m];
        h1[orow * 16 + m] = acc;
      }
    }
  }
  (void)row;
}

// ---------------------------------------------------------------------------
// layer-1 scatter: one thread per (edge, quarter)
// 1x global_load_b128 gather + 4 f32 atomics per thread; 4 threads per edge
// agg1[dst][4q..4q+3] += h1[src][4q..4q+3] * dinv[src] * dinv[dst]
// ---------------------------------------------------------------------------
__global__ __launch_bounds__(256) void scatter1_kernel(const long long* __restrict__ src,
                                                       const long long* __restrict__ dst,
                                                       const float* __restrict__ dinv,
                                                       const float* __restrict__ h1,
                                                       float* __restrict__ agg1,
                                                       long long E4) {
  long long idx = (long long)blockIdx.x * blockDim.x + threadIdx.x;
  if (idx >= E4) return;
  int e = (int)(idx >> 2);
  int q = (int)(idx & 3);
  int s = (int)src[e];
  int d = (int)dst[e];
  float nrm = dinv[s] * dinv[d];
  const float4 v = *(const float4*)(h1 + (size_t)s * 16 + q * 4);
  float* ap = agg1 + (size_t)d * 16 + q * 4;
  atomicAdd(ap + 0, v.x * nrm);
  atomicAdd(ap + 1, v.y * nrm);
  atomicAdd(ap + 2, v.z * nrm);
  atomicAdd(ap + 3, v.w * nrm);
}

// ---------------------------------------------------------------------------
// finalize layer 1 (in place, float4): agg1 = relu(agg1 + self_loop + b1)
// grid covers N*4 float4 slots
// ---------------------------------------------------------------------------
__global__ __launch_bounds__(256) void finalize1_kernel(const float4* __restrict__ h1,
                                                        const float* __restrict__ dinv,
                                                        const float4* __restrict__ b1,
                                                        float4* __restrict__ agg1, int N4) {
  int i = blockIdx.x * blockDim.x + threadIdx.x;
  if (i >= N4) return;
  int n = i >> 2;
  int q = i & 3;
  float di = dinv[n];
  float sl = di * di;
  float4 a = agg1[i];
  float4 h = h1[i];
  float4 b = b1[q];
  float4 v;
  v.x = fmaxf(a.x + h.x * sl + b.x, 0.0f);
  v.y = fmaxf(a.y + h.y * sl + b.y, 0.0f);
  v.z = fmaxf(a.z + h.z * sl + b.z, 0.0f);
  v.w = fmaxf(a.w + h.w * sl + b.w, 0.0f);
  agg1[i] = v;
}

// ---------------------------------------------------------------------------
// t = h2 @ W2   (N x 16) @ (16 x 1) via 4 accumulating 16x16x4 WMMAs
// (B has only column 0 populated; lanes with (lane&15)==0 hold the result)
// ---------------------------------------------------------------------------
__global__ __launch_bounds__(256) void dot2_wmma_kernel(const float* __restrict__ h2,
                                                        const float* __restrict__ W2,
                                                        float* __restrict__ t, int N) {
  int wave = (blockIdx.x * blockDim.x + threadIdx.x) >> 5;
  int lane = threadIdx.x & 31;
  int row0 = wave * 16;
  if (row0 >= N) return;            // uniform per wave

  int m  = lane & 15;
  int kh = lane >> 4;
  int row = row0 + m;
  bool full = (row0 + 16 <= N);     // wave-uniform

  if (full) {
#if HAS_WMMA_F32X4
    v8f c = {};
#pragma unroll
    for (int kk = 0; kk < 4; ++kk) {
      int k0 = kk * 4 + 2 * kh;                    // 0..14, even
      // 8B-aligned pair load of A operand
      v2f a = *(const v2f*)(h2 + (size_t)row * 16 + k0);
      // W2 loaded unconditionally (always in bounds), masked via cndmask
      float w0 = W2[k0];
      float w1 = W2[k0 + 1];
      v2f b;
      b.x = (m == 0) ? w0 : 0.0f;
      b.y = (m == 0) ? w1 : 0.0f;
      c = __builtin_amdgcn_wmma_f32_16x16x4_f32(false, a, false, b, (short)0, c,
                                                false, false);
    }
    if (m == 0) {
#pragma unroll
      for (int r = 0; r < 8; ++r)
        t[row0 + r + 8 * kh] = c[r];
    }
#else
    if (m == 0) {
#pragma unroll
      for (int r = 0; r < 8; ++r) {
        int orow = row0 + r + 8 * kh;
        float acc = 0.0f;
#pragma unroll
        for (int j = 0; j < 16; ++j) acc += h2[orow * 16 + j] * W2[j];
        t[orow] = acc;
      }
    }
#endif
  } else {
    if (m == 0) {
#pragma unroll
      for (int r = 0; r < 8; ++r) {
        int orow = row0 + r + 8 * kh;
        if (orow < N) {
          float acc = 0.0f;
#pragma unroll
          for (int j = 0; j < 16; ++j) acc += h2[orow * 16 + j] * W2[j];
          t[orow] = acc;
        }
      }
    }
  }
  (void)row;
}

// ---------------------------------------------------------------------------
// layer-2 scatter: one thread per edge
// ---------------------------------------------------------------------------
__global__ __launch_bounds__(256) void scatter2_kernel(const long long* __restrict__ src,
                                                       const long long* __restrict__ dst,
                                                       const float* __restrict__ dinv,
                                                       const float* __restrict__ t,
                                                       float* __restrict__ agg2, int E) {
  int e = blockIdx.x * blockDim.x + threadIdx.x;
  if (e >= E) return;
  int s = (int)src[e];
  int d = (int)dst[e];
  atomicAdd(&agg2[d], t[s] * dinv[s] * dinv[d]);
}

// ---------------------------------------------------------------------------
// finalize layer 2: out = agg2 + self_loop + b2
// ---------------------------------------------------------------------------
__global__ __launch_bounds__(256) void finalize2_kernel(const float* __restrict__ t,
                                                        const float* __restrict__ dinv,
                                                        const float* __restrict__ b2,
                                                        const float* __restrict__ agg2,
                                                        float* __restrict__ out, int N) {
  int n = blockIdx.x * blockDim.x + threadIdx.x;
  if (n >= N) return;
  float di = dinv[n];
  out[n] = agg2[n] + t[n] * di * di + b2[0];
}

// ---------------------------------------------------------------------------
extern "C" void kernel_launch(void* const* d_in, const int* in_sizes, int n_in,
                              void* d_out, int out_size, void* d_ws, size_t ws_size,
                              hipStream_t stream) {
  const float*     x  = (const float*)d_in[0];
  const long long* ei = (const long long*)d_in[1];   // int64 [2, E] flat
  const float*     W1 = (const float*)d_in[2];       // [3,16]
  const float*     b1 = (const float*)d_in[3];       // [16]
  const float*     W2 = (const float*)d_in[4];       // [16,1]
  const float*     b2 = (const float*)d_in[5];       // [1]

  const int N = in_sizes[0] / 3;
  const int E = in_sizes[1] / 2;
  const long long* src = ei;
  const long long* dst = ei + E;

  // workspace layout: dinv[N] | h1[16N] | agg1[16N] (reused as h2) | t[N] | agg2[N]
  float* ws   = (float*)d_ws;
  float* dinv = ws;
  float* h1   = dinv + N;
  float* agg1 = h1 + (size_t)16 * N;
  float* t    = agg1 + (size_t)16 * N;
  float* agg2 = t + N;
  float* out  = (float*)d_out;

  const int thr = 256;
  const int N4 = N * 4;                       // float4 slots in agg1
  const long long E4 = (long long)E * 4;      // (edge, quarter) threads

  init_ws_kernel<<<(N4 + thr - 1) / thr, thr, 0, stream>>>(dinv, (float4*)agg1, agg2, N);
  deg_accum_kernel<<<(E + thr - 1) / thr, thr, 0, stream>>>(dst, dinv, E);
  dinv_kernel<<<(N + thr - 1) / thr, thr, 0, stream>>>(dinv, N);

  const int tiles = (N + 15) / 16;
  const int wpb = thr / 32;                   // waves per block (wave32)
  h1_wmma_kernel<<<(tiles + wpb - 1) / wpb, thr, 0, stream>>>(x, W1, h1, N);

  scatter1_kernel<<<(unsigned)((E4 + thr - 1) / thr), thr, 0, stream>>>(src, dst, dinv, h1, agg1, E4);
  finalize1_kernel<<<(N4 + thr - 1) / thr, thr, 0, stream>>>((const float4*)h1, dinv,
                                                             (const float4*)b1, (float4*)agg1, N4);

  dot2_wmma_kernel<<<(tiles + wpb - 1) / wpb, thr, 0, stream>>>(agg1, W2, t, N);

  scatter2_kernel<<<(E + thr - 1) / thr, thr, 0, stream>>>(src, dst, dinv, t, agg2, E);
  finalize2_kernel<<<(N + thr - 1) / thr, thr, 0, stream>>>(t, dinv, b2, agg2, out, N);
}